// GCN_35244501631243
// MI455X (gfx1250) — compile-verified
//
#include <hip/hip_runtime.h>
#include <hip/hip_bf16.h>

#define DFEAT 96

typedef float v2f __attribute__((ext_vector_type(2)));
typedef float v8f __attribute__((ext_vector_type(8)));

// ---------------- elementwise / setup kernels ----------------

__global__ void k_fill(float* __restrict__ p, float val, int n) {
  int i = blockIdx.x * blockDim.x + threadIdx.x;
  if (i < n) p[i] = val;
}

__global__ void k_deg_accum(const int* __restrict__ dst, const float* __restrict__ ew,
                            float* __restrict__ deg, int ne) {
  int e = blockIdx.x * blockDim.x + threadIdx.x;
  if (e < ne) atomicAdd(&deg[dst[e]], ew[e]);
}

__global__ void k_dinv(const float* __restrict__ deg, float* __restrict__ dinv, int n) {
  int i = blockIdx.x * blockDim.x + threadIdx.x;
  if (i < n) {
    float d = deg[i];
    dinv[i] = (d > 0.0f) ? rsqrtf(fmaxf(d, 1e-20f)) : 0.0f;
  }
}

// ------------- FP32 WMMA GEMM with fused self-loop/bias epilogue -------------
//   T[n,96]   = act(X)[n,96] @ W[96,96]            (gathered by edge scatter)
//   out[n,96] = dinv[row]^2 * T + bias[col]        (self-loop term + bias;
//                                                   scatter atomics add on top)
// act = relu if relu_in != 0 (defers layer-1 relu into layer-3/4 A loads).
// One wave per 16-row M-tile; six 16-col N-tiles in v8f accumulators;
// K=96 in 24 steps of v_wmma_f32_16x16x4_f32.
// W staged in LDS interleaved as float2 {W[kb][col], W[kb+1][col]} so each
// B fragment is one aligned ds_load_b64 directly into the WMMA source pair.
__global__ __launch_bounds__(256) void k_gemm_wmma_fused(
    const float* __restrict__ X, const float* __restrict__ W,
    const float* __restrict__ dinv, const float* __restrict__ bias,
    float* __restrict__ T, float* __restrict__ out,
    int nrows, int relu_in) {
  __shared__ v2f sW2[(DFEAT / 2) * DFEAT];  // 36 KB; index (kb/2)*96 + col
  for (int idx = threadIdx.x; idx < (DFEAT / 2) * DFEAT; idx += 256) {
    const int kp  = idx / DFEAT;            // K pair 0..47
    const int col = idx - kp * DFEAT;
    v2f w;
    w.x = W[(2 * kp) * DFEAT + col];
    w.y = W[(2 * kp + 1) * DFEAT + col];
    sW2[idx] = w;
  }
  __syncthreads();

  const int wave = threadIdx.x >> 5;
  const int lane = threadIdx.x & 31;
  const int ntiles = nrows >> 4;                // N = 50000 divisible by 16
  const int mt = blockIdx.x * 8 + wave;         // wave-uniform -> EXEC all-1s for WMMA
  if (mt >= ntiles) return;

  const int lo16  = lane & 15;
  const int khalf = lane >> 4;                  // 0: K={kb,kb+1}  1: K={kb+2,kb+3}
  const int row   = mt * 16 + lo16;
  const float* xr = X + (size_t)row * DFEAT;

  v8f acc[6];
#pragma unroll
  for (int t = 0; t < 6; ++t) acc[t] = (v8f){0.f, 0.f, 0.f, 0.f, 0.f, 0.f, 0.f, 0.f};

#pragma unroll 4
  for (int k = 0; k < 24; ++k) {
    const int kb = 4 * k + 2 * khalf;           // even -> 8B aligned
    v2f a = *(const v2f*)(xr + kb);
    if (relu_in) {
      a.x = (a.x > 0.0f) ? a.x : 0.0f;
      a.y = (a.y > 0.0f) ? a.y : 0.0f;
    }
    const v2f* brow = sW2 + (kb >> 1) * DFEAT + lo16;   // {W[kb][c], W[kb+1][c]}
#pragma unroll
    for (int t = 0; t < 6; ++t) {
      v2f b = brow[t * 16];                      // single ds_load_b64, even pair
      acc[t] = __builtin_amdgcn_wmma_f32_16x16x4_f32(
          /*neg_a=*/false, a, /*neg_b=*/false, b,
          /*c_mod=*/(short)0, acc[t], /*reuse_a=*/false, /*reuse_b=*/false);
    }
  }

  // C/D layout: VGPR r -> M=r (lanes 0-15) / M=r+8 (lanes 16-31); N = lane&15
  const int rbase = mt * 16 + khalf * 8;
  float dv2[8];
#pragma unroll
  for (int r = 0; r < 8; ++r) {
    float d = dinv[rbase + r];
    dv2[r] = d * d;
  }
#pragma unroll
  for (int t = 0; t < 6; ++t) {
    const int cbase = t * 16 + lo16;
    const float bb = bias[cbase];
#pragma unroll
    for (int r = 0; r < 8; ++r) {
      const size_t idx = (size_t)(rbase + r) * DFEAT + cbase;
      const float v = acc[t][r];
      T[idx] = v;
      out[idx] = dv2[r] * v + bb;
    }
  }
}

// ---------------- edge scatter:  out[dst] += dinv[src]*ew*dinv[dst] * T[src] --------
// One wave per edge; 3 features per lane (96 = 3*32), coalesced rows, f32 atomics.
__global__ __launch_bounds__(256) void k_scatter(const float* __restrict__ T,
                                                 const int* __restrict__ src,
                                                 const int* __restrict__ dst,
                                                 const float* __restrict__ ew,
                                                 const float* __restrict__ dinv,
                                                 float* __restrict__ out, int ne) {
  const int e = blockIdx.x * 8 + (threadIdx.x >> 5);
  if (e >= ne) return;
  const int lane = threadIdx.x & 31;
  const int s = src[e];
  const int d = dst[e];
  const float nrm = dinv[s] * ew[e] * dinv[d];
  const float* ts = T + (size_t)s * DFEAT;
  float* od = out + (size_t)d * DFEAT;
#pragma unroll
  for (int j = 0; j < 3; ++j) {
    const int f = lane + 32 * j;
    atomicAdd(&od[f], nrm * ts[f]);
  }
}

// ---------------- global_add_pool ----------------
__global__ void k_pool(const float* __restrict__ ge, const int* __restrict__ batch,
                       float* __restrict__ gout, int total) {
  int i = blockIdx.x * blockDim.x + threadIdx.x;
  if (i >= total) return;
  int node = i / DFEAT;
  int f = i - node * DFEAT;
  atomicAdd(&gout[batch[node] * DFEAT + f], ge[i]);
}

// ---------------- launcher ----------------
extern "C" void kernel_launch(void* const* d_in, const int* in_sizes, int n_in,
                              void* d_out, int out_size, void* d_ws, size_t ws_size,
                              hipStream_t stream) {
  (void)n_in; (void)ws_size;
  const float* x    = (const float*)d_in[0];
  const float* W1   = (const float*)d_in[1];
  const float* b1   = (const float*)d_in[2];
  const float* W3   = (const float*)d_in[3];
  const float* b3   = (const float*)d_in[4];
  const float* W4   = (const float*)d_in[5];
  const float* b4   = (const float*)d_in[6];
  const float* ew   = (const float*)d_in[7];
  const int*   ei   = (const int*)d_in[8];
  const int*   batch= (const int*)d_in[9];

  const int n  = in_sizes[0] / DFEAT;   // 50000
  const int ne = in_sizes[7];           // 800000
  const int* srcp = ei;                 // edge_index row 0
  const int* dstp = ei + ne;            // edge_index row 1

  const int nf = n * DFEAT;             // 4.8M
  const int gtotal = out_size - nf;     // G*DFEAT

  // workspace layout (floats): deg | dinv | tmp(GEMM out) | h | tge  (~58 MB)
  float* deg  = (float*)d_ws;
  float* dinv = deg + n;
  float* tmp  = dinv + n;
  float* h    = tmp + (size_t)nf;       // pre-relu hidden; relu fused into A loads
  float* tge  = h + (size_t)nf;

  float* embed = (float*)d_out;         // [n, 96]
  float* gout  = embed + nf;            // [G, 96]

  const dim3 B(256);
  const int gN  = (n + 255) / 256;
  const int gE  = (ne + 255) / 256;
  const int gF  = (nf + 255) / 256;
  const int gSc = (ne + 7) / 8;         // 8 edges (waves) per block
  const int gMM = ((n >> 4) + 7) / 8;   // 8 M-tiles (waves) per block

  // shared symmetric normalization (identical for all three convs)
  k_fill<<<gN, B, 0, stream>>>(deg, 1.0f, n);                       // self-loop weight
  k_deg_accum<<<gE, B, 0, stream>>>(dstp, ew, deg, ne);
  k_dinv<<<gN, B, 0, stream>>>(deg, dinv, n);

  // layer 1: h_pre = A_hat @ (x@W1) + b1   (relu deferred into layer-3/4 loads)
  k_gemm_wmma_fused<<<gMM, B, 0, stream>>>(x, W1, dinv, b1, tmp, h, n, 0);
  k_scatter<<<gSc, B, 0, stream>>>(tmp, srcp, dstp, ew, dinv, h, ne);

  // layer 3: embed = A_hat @ (relu(h_pre)@W3) + b3   (straight into d_out)
  k_gemm_wmma_fused<<<gMM, B, 0, stream>>>(h, W3, dinv, b3, tmp, embed, n, 1);
  k_scatter<<<gSc, B, 0, stream>>>(tmp, srcp, dstp, ew, dinv, embed, ne);

  // layer 4 + pool: graph_embed = segment_sum(A_hat @ (relu(h_pre)@W4) + b4, batch)
  k_gemm_wmma_fused<<<gMM, B, 0, stream>>>(h, W4, dinv, b4, tmp, tge, n, 1);
  k_scatter<<<gSc, B, 0, stream>>>(tmp, srcp, dstp, ew, dinv, tge, ne);
  k_fill<<<(gtotal + 255) / 256, B, 0, stream>>>(gout, 0.0f, gtotal);
  k_pool<<<gF, B, 0, stream>>>(tge, batch, gout, nf);
}